// AR_10746008175387
// MI455X (gfx1250) — compile-verified
//
#include <hip/hip_runtime.h>

typedef float v2f __attribute__((ext_vector_type(2)));
typedef float v8f __attribute__((ext_vector_type(8)));

#define B_TOT  16384
#define D_DIM  64
#define H_DIM  1024
#define PD     128
#define ROWS   32      // samples per block (2 waves x 16)
#define AS     1028    // padded LDS row stride for 'a' (bank-conflict-free WMMA A loads)
#define THREADS 64

// ---- one-time weight pack: sort hidden units by m_hid, transpose W2 ----
// sorted index j -> original h:  m_hid values 1..16 appear 17x, 17..63 appear 16x
//   j < 272 : v = j/17 + 1, k = j%17      (S(17) = 17*16 = 272)
//   j >= 272: v = 17 + (j-272)/16, k = (j-272)%16
//   h = (v-1) + 63*k
// Prefix count: cnt(d) = #{j : m(j) <= d} = 16*d + min(d,16)
__global__ __launch_bounds__(256)
void pack_weights(const float* __restrict__ W1, const float* __restrict__ b1,
                  const float* __restrict__ W2,
                  float* __restrict__ W2Ts, float* __restrict__ W1s,
                  float* __restrict__ b1s)
{
    int j = blockIdx.x * 256 + threadIdx.x;
    if (j >= H_DIM) return;
    int v, k;
    if (j < 17 * 16) { v = j / 17 + 1;            k = j % 17; }
    else             { v = 17 + (j - 272) / 16;   k = (j - 272) % 16; }
    int h = (v - 1) + 63 * k;
    b1s[j] = b1[h];
    for (int c = 0; c < PD; ++c)
        W2Ts[c * H_DIM + j] = W2[h * PD + c];       // transpose + permute rows
    for (int d = 0; d < D_DIM; ++d)
        W1s[d * H_DIM + j] = W1[d * H_DIM + h];     // permute columns
}

__global__ __launch_bounds__(THREADS)
void made_flow_wmma(const float* __restrict__ z,
                    const int*   __restrict__ perm,
                    const float* __restrict__ W1s,
                    const float* __restrict__ b1s,
                    const float* __restrict__ W2Ts,
                    const float* __restrict__ b2,
                    float* __restrict__ out)
{
    extern __shared__ float smem[];
    float* a    = smem;                 // ROWS * AS   (pre-activations, sorted hidden order)
    float* w1m  = a + ROWS * AS;        // H           (masked W1s row d: live for j >= cnt)
    float* cols = w1m + H_DIM;          // H           (masked W2Ts[d]:   live for j <  cnt)
    float* colt = cols + H_DIM;         // H           (masked W2Ts[64+d])
    float* zbuf = colt + H_DIM;         // H           (static zeros: B operand, dead N-lanes)
    float* sbuf = zbuf + H_DIM;         // ROWS
    float* tbuf = sbuf + ROWS;          // ROWS
    float* xbuf = tbuf + ROWS;          // ROWS

    const int tid    = threadIdx.x;
    const int lane   = tid & 31;
    const int wv     = tid >> 5;
    const int row    = lane & 15;       // WMMA A: M index / B,C: N index
    const int hi     = lane >> 4;       // half-wave selector
    const int base_m = wv * 16;
    const int s0     = blockIdx.x * ROWS;

    for (int h = tid; h < H_DIM; h += THREADS) zbuf[h] = 0.f;
    for (int idx = tid; idx < ROWS * H_DIM; idx += THREADS) {
        int m = idx >> 10, h = idx & (H_DIM - 1);
        a[m * AS + h] = b1s[h];
    }
    float logdet = 0.f;
    __syncthreads();

    // per-lane B-operand source: N==0 -> s column, N==1 -> t column, else zeros
    const float* bsel = (row == 0) ? cols : (row == 1) ? colt : zbuf;
    const v2f*   bp2  = (const v2f*)bsel;

    for (int d = 0; d < D_DIM; ++d) {
        const int cnt   = 16 * d + (d < 16 ? d : 16);  // boundary in sorted order
        const int kcpad = (cnt + 15) & ~15;            // WMMA K extent (16-aligned up)
        const int plo   = cnt & ~15;                   // rank-1 start  (16-aligned down)

        // ---- staging: only the live prefix/suffix is touched ----
        for (int j = tid; j < kcpad; j += THREADS) {
            float keep = (j < cnt) ? 1.f : 0.f;
            cols[j] = keep * W2Ts[d * H_DIM + j];
            colt[j] = keep * W2Ts[(D_DIM + d) * H_DIM + j];
        }
        for (int j = plo + tid; j < H_DIM; j += THREADS)
            w1m[j] = (j >= cnt) ? W1s[d * H_DIM + j] : 0.f;
        if (d + 1 < D_DIM) {
            __builtin_prefetch(&W1s[(d + 1) * H_DIM + tid * 16], 0, 0);
            __builtin_prefetch(&W2Ts[(d + 1) * H_DIM + tid * 16], 0, 0);
            __builtin_prefetch(&W2Ts[(D_DIM + d + 1) * H_DIM + tid * 16], 0, 0);
        }
        __syncthreads();

        // ---- s,t = relu(a[16 x cnt]) @ [cols|colt|0...] via V_WMMA_F32_16X16X4_F32 ----
        v8f acc0 = {0.f,0.f,0.f,0.f,0.f,0.f,0.f,0.f};
        v8f acc1 = acc0, acc2 = acc0, acc3 = acc0;
        const v2f* arow2 = (const v2f*)&a[(base_m + row) * AS];   // 8B-aligned (4112*m)
        #pragma unroll 2
        for (int kc = 0; kc < (kcpad >> 2); kc += 4) {
            int i = kc * 2 + hi;                 // A layout: K = vgpr + 2*(lane>=16)
            v2f a0 = arow2[i],     a1 = arow2[i + 2];
            v2f a2 = arow2[i + 4], a3 = arow2[i + 6];
            v2f b0 = bp2[i],       b1v = bp2[i + 2];
            v2f b2v = bp2[i + 4],  b3 = bp2[i + 6];
            a0.x = (a0.x > 0.f) ? a0.x : 0.f;  a0.y = (a0.y > 0.f) ? a0.y : 0.f;
            a1.x = (a1.x > 0.f) ? a1.x : 0.f;  a1.y = (a1.y > 0.f) ? a1.y : 0.f;
            a2.x = (a2.x > 0.f) ? a2.x : 0.f;  a2.y = (a2.y > 0.f) ? a2.y : 0.f;
            a3.x = (a3.x > 0.f) ? a3.x : 0.f;  a3.y = (a3.y > 0.f) ? a3.y : 0.f;
            acc0 = __builtin_amdgcn_wmma_f32_16x16x4_f32(false, a0, false, b0,
                                                         (short)0, acc0, false, false);
            acc1 = __builtin_amdgcn_wmma_f32_16x16x4_f32(false, a1, false, b1v,
                                                         (short)0, acc1, false, false);
            acc2 = __builtin_amdgcn_wmma_f32_16x16x4_f32(false, a2, false, b2v,
                                                         (short)0, acc2, false, false);
            acc3 = __builtin_amdgcn_wmma_f32_16x16x4_f32(false, a3, false, b3,
                                                         (short)0, acc3, false, false);
        }
        v8f acc = (acc0 + acc1) + (acc2 + acc3);

        // C layout: lane holds column N=row, rows M = r + 8*hi
        if (row == 0) {
            #pragma unroll
            for (int r = 0; r < 8; ++r) sbuf[base_m + 8 * hi + r] = acc[r];
        } else if (row == 1) {
            #pragma unroll
            for (int r = 0; r < 8; ++r) tbuf[base_m + 8 * hi + r] = acc[r];
        }
        __syncthreads();

        // ---- sequential autoregressive update (one thread per sample) ----
        if (tid < ROWS) {
            int   sample = s0 + tid;
            float s  = sbuf[tid] + b2[d];
            float t  = tbuf[tid] + b2[D_DIM + d];
            logdet  += s;
            int   pd = perm[d];
            float xv = z[sample * D_DIM + pd] * expf(s) + t;
            out[sample * D_DIM + pd] = xv;   // scatter == x[:, argsort(perm)] gather
            xbuf[tid] = xv;
        }
        __syncthreads();

        // ---- rank-1 update on live suffix: a[m][j] += x_m * w1m[j], j in [plo, H) ----
        {
            int m    = tid >> 1;
            int half = tid & 1;
            float xm = xbuf[m];
            float4*       ap = (float4*)&a[m * AS];
            const float4* wp = (const float4*)w1m;
            #pragma unroll 4
            for (int i = (plo >> 2) + half; i < H_DIM / 4; i += 2) {
                float4 av4 = ap[i];
                float4 wv4 = wp[i];
                av4.x += xm * wv4.x; av4.y += xm * wv4.y;
                av4.z += xm * wv4.z; av4.w += xm * wv4.w;
                ap[i] = av4;
            }
        }
        __syncthreads();
    }

    if (tid < ROWS)
        out[B_TOT * D_DIM + s0 + tid] = logdet;
}

extern "C" void kernel_launch(void* const* d_in, const int* in_sizes, int n_in,
                              void* d_out, int out_size, void* d_ws, size_t ws_size,
                              hipStream_t stream) {
    const float* z    = (const float*)d_in[0];
    const int*   perm = (const int*)  d_in[1];
    const float* W1   = (const float*)d_in[2];
    const float* b1   = (const float*)d_in[3];
    const float* W2   = (const float*)d_in[4];
    const float* b2   = (const float*)d_in[5];
    float*       out  = (float*)d_out;

    float* W2Ts = (float*)d_ws;              // PD*H   floats (512 KB)
    float* W1s  = W2Ts + (size_t)PD * H_DIM; // D*H    floats (256 KB)
    float* b1s  = W1s + (size_t)D_DIM * H_DIM; // H    floats (4 KB)

    hipLaunchKernelGGL(pack_weights, dim3(H_DIM / 256), dim3(256), 0, stream,
                       W1, b1, W2, W2Ts, W1s, b1s);

    size_t smem = (size_t)(ROWS * AS + 4 * H_DIM + 3 * ROWS) * sizeof(float);
    hipLaunchKernelGGL(made_flow_wmma, dim3(B_TOT / ROWS), dim3(THREADS), smem, stream,
                       z, perm, W1s, b1s, W2Ts, b2, out);
}